// RnnModel_47888885350930
// MI455X (gfx1250) — compile-verified
//
#include <hip/hip_runtime.h>
#include <hip/hip_bf16.h>

typedef __attribute__((ext_vector_type(2))) float v2f;
typedef __attribute__((ext_vector_type(8))) float v8f;

#define H 50
#define TT 512
#define LDW 64   // padded row width for pre/ys buffers

__device__ __forceinline__ float fast_tanh(float x) {
#if __has_builtin(__builtin_amdgcn_tanhf)
  return __builtin_amdgcn_tanhf(x);    // V_TANH_F32 (gfx1250 TRANS op)
#else
  return tanhf(x);
#endif
}

// ---------------------------------------------------------------------------
// Prep: Wpad[64][Kpad] = W (zero-padded rows>=H, cols>=Din);
//       biasp[64] = b_ih + b_hh (zero-padded).
// Padding guarantees the WMMA GEMM needs no bounds checks and writes exact
// zeros into pre[:, H..LDW), which in turn makes the scan branch-free.
// ---------------------------------------------------------------------------
__global__ __launch_bounds__(256) void pad_weights(
    const float* __restrict__ W, const float* __restrict__ bih,
    const float* __restrict__ bhh, int Din, int Kpad,
    float* __restrict__ Wpad, float* __restrict__ biasp) {
  const int idx = blockIdx.x * blockDim.x + threadIdx.x;
  const int total = LDW * Kpad;
  if (idx < total) {
    const int r = idx / Kpad;
    const int c = idx % Kpad;
    Wpad[idx] = (r < H && c < Din) ? W[r * Din + c] : 0.f;
  }
  if (idx < LDW) biasp[idx] = (idx < H) ? (bih[idx] + bhh[idx]) : 0.f;
}

// ---------------------------------------------------------------------------
// pre[t][h] = sum_d X[t][d] * Wpad[h][d] + biasp[h], via V_WMMA_F32_16X16X4_F32.
// Grid: 32 blocks x 4 waves; block = 16 time rows, wave w = output cols [16w,16w+16).
// A (16x4 f32): lanes 0-15 hold {K,K+1} of row M=lane; lanes 16-31 hold {K+2,K+3}.
// B (4x16 f32): same pairing with N=lane (B = Wpad^T, Wpad is N-major).
// C/D: VGPR r -> row r (lanes 0-15) / r+8 (lanes 16-31), N = lane%16.
// Branch-free K loop: unconditional 8B float2 loads, 2 accumulators, unroll x2.
// ---------------------------------------------------------------------------
__global__ __launch_bounds__(128) void wmma_pre_gemm(
    const float* __restrict__ X, long long xRowStride, int Kpad,
    const float* __restrict__ Wpad /* LDW x Kpad */,
    const float* __restrict__ biasp /* LDW */,
    float* __restrict__ out /* TT x LDW */) {
  const int t0   = blockIdx.x * 16;
  const int wave = threadIdx.x >> 5;
  const int lane = threadIdx.x & 31;
  const int lm   = lane & 15;
  const int hi   = lane >> 4;          // 0: K pair {0,1}, 1: K pair {2,3}
  const int h0   = wave * 16;

  const int rowA = t0 + lm;            // time row (M)
  const int colB = h0 + lm;            // hidden col (N) == Wpad row

  const v2f* __restrict__ xr = (const v2f*)(X + (long long)rowA * xRowStride);
  const v2f* __restrict__ wr = (const v2f*)(Wpad + (long long)colB * Kpad);

  v8f c0 = {}, c1 = {};
  const int n2 = Kpad >> 1;            // float2 elements per row
  int k2 = 0;
  for (; k2 + 4 <= n2; k2 += 4) {      // K += 8 per iteration, 2 indep. chains
    const v2f a0 = xr[k2 + hi];
    const v2f b0 = wr[k2 + hi];
    const v2f a1 = xr[k2 + 2 + hi];
    const v2f b1 = wr[k2 + 2 + hi];
    c0 = __builtin_amdgcn_wmma_f32_16x16x4_f32(false, a0, false, b0,
                                               (short)0, c0, false, false);
    c1 = __builtin_amdgcn_wmma_f32_16x16x4_f32(false, a1, false, b1,
                                               (short)0, c1, false, false);
  }
  for (; k2 < n2; k2 += 2) {           // K remainder (Kpad % 8)
    const v2f a = xr[k2 + hi];
    const v2f b = wr[k2 + hi];
    c0 = __builtin_amdgcn_wmma_f32_16x16x4_f32(false, a, false, b,
                                               (short)0, c0, false, false);
  }

  const float bias = biasp[colB];      // exact 0 for colB >= H
#pragma unroll
  for (int r = 0; r < 8; ++r) {
    const int row = t0 + r + hi * 8;
    out[row * LDW + h0 + lm] = (c0[r] + c1[r]) + bias;
  }
}

// ---------------------------------------------------------------------------
// Sequential recurrence: h = tanh(pre[t] + h @ W_hh^T), ys[t] = h.
// One workgroup, 2 waves, lane i owns element i.  pre[:,H..LDW) is exact zero
// (padded GEMM) -> fully unconditional loads/stores; pad lanes compute
// tanh(0)=0, keeping ys padding columns zero for the next layer's A loads.
// h ping-pongs between two LDS buffers so only ONE barrier per step sits on
// the 512-step dependency chain: reads of hs[t&1] happen before the barrier,
// next step's writes to hs[t&1] happen after it.
// W_hh row in VGPRs; h broadcast via ds_load_b128; 4 independent FMA chains.
// ---------------------------------------------------------------------------
__global__ __launch_bounds__(64) void rnn_scan(
    const float* __restrict__ pre /* TT x LDW */,
    const float* __restrict__ Whh /* H x H */,
    float* __restrict__ ys /* TT x LDW */, int T) {
  __shared__ __align__(16) float hs[2][LDW];
  const int i = threadIdx.x;           // 0..63

  float Wrow[52];
#pragma unroll
  for (int j = 0; j < 52; ++j) Wrow[j] = 0.f;
  if (i < H) {
#pragma unroll
    for (int j = 0; j < H; ++j) Wrow[j] = Whh[i * H + j];
  }

  hs[0][i] = 0.f;
  __syncthreads();

  float nxt = pre[i];                  // software prefetch of pre[t] row
  for (int t = 0; t < T; ++t) {
    float acc0 = nxt, acc1 = 0.f, acc2 = 0.f, acc3 = 0.f;
    const int tn = (t + 1 < T) ? (t + 1) : t;   // clamp: branch-free prefetch
    nxt = pre[tn * LDW + i];

    const float4* __restrict__ h4 = (const float4*)hs[t & 1];
#pragma unroll
    for (int c2 = 0; c2 < 13; ++c2) {  // 52 MACs, 4 chains of depth 13
      const float4 v = h4[c2];
      acc0 = fmaf(Wrow[4 * c2 + 0], v.x, acc0);
      acc1 = fmaf(Wrow[4 * c2 + 1], v.y, acc1);
      acc2 = fmaf(Wrow[4 * c2 + 2], v.z, acc2);
      acc3 = fmaf(Wrow[4 * c2 + 3], v.w, acc3);
    }
    const float hn = fast_tanh((acc0 + acc1) + (acc2 + acc3));
    hs[(t + 1) & 1][i] = hn;           // other buffer: no conflict with reads
    ys[t * LDW + i] = hn;              // pad lanes write exact 0
    __syncthreads();                   // single barrier on the critical path
  }
}

// ---------------------------------------------------------------------------
// out[t][o] = sum_h ys[t][h] * W_lin[o][h] + b_lin[o]   (512 x 5)
// ---------------------------------------------------------------------------
__global__ __launch_bounds__(128) void final_linear(
    const float* __restrict__ ys /* TT x LDW */,
    const float* __restrict__ Wl /* 5 x H */,
    const float* __restrict__ bl /* 5 */,
    float* __restrict__ out /* TT x 5 */) {
  const int idx = blockIdx.x * blockDim.x + threadIdx.x;
  if (idx >= TT * 5) return;
  const int t = idx / 5;
  const int o = idx % 5;
  float acc = bl[o];
#pragma unroll
  for (int h = 0; h < H; ++h) acc = fmaf(ys[t * LDW + h], Wl[o * H + h], acc);
  out[t * 5 + o] = acc;
}

extern "C" void kernel_launch(void* const* d_in, const int* in_sizes, int n_in,
                              void* d_out, int out_size, void* d_ws, size_t ws_size,
                              hipStream_t stream) {
  const float* x     = (const float*)d_in[0];   // (512, 256, 300)
  const float* W_ih0 = (const float*)d_in[1];
  const float* W_hh0 = (const float*)d_in[2];
  const float* b_ih0 = (const float*)d_in[3];
  const float* b_hh0 = (const float*)d_in[4];
  const float* W_ih1 = (const float*)d_in[5];
  const float* W_hh1 = (const float*)d_in[6];
  const float* b_ih1 = (const float*)d_in[7];
  const float* b_hh1 = (const float*)d_in[8];
  const float* W_ih2 = (const float*)d_in[9];
  const float* W_hh2 = (const float*)d_in[10];
  const float* b_ih2 = (const float*)d_in[11];
  const float* b_hh2 = (const float*)d_in[12];
  const float* W_lin = (const float*)d_in[13];
  const float* b_lin = (const float*)d_in[14];

  float* bufA  = (float*)d_ws;          // pre   (512 x 64)
  float* bufB  = bufA + TT * LDW;       // ys    (512 x 64)
  float* Wpad  = bufB + TT * LDW;       // up to 64 x 304 padded weights
  float* biasp = Wpad + LDW * 304;      // 64 padded bias

  // Only batch element 255 reaches the output (h[:, -1, :] indexes batch).
  const float*    x255    = x + 255 * 300;
  const long long xstride = 256LL * 300LL;

  const int K0 = 300;                   // layer-0 K (multiple of 4)
  const int K1 = 52;                    // layers 1/2: H=50 padded to 52

  // Layer 0
  pad_weights  <<<(LDW * K0 + 255) / 256, 256, 0, stream>>>(W_ih0, b_ih0, b_hh0, 300, K0, Wpad, biasp);
  wmma_pre_gemm<<<32, 128, 0, stream>>>(x255, xstride, K0, Wpad, biasp, bufA);
  rnn_scan     <<<1, 64, 0, stream>>>(bufA, W_hh0, bufB, TT);
  // Layer 1
  pad_weights  <<<(LDW * K1 + 255) / 256, 256, 0, stream>>>(W_ih1, b_ih1, b_hh1, H, K1, Wpad, biasp);
  wmma_pre_gemm<<<32, 128, 0, stream>>>(bufB, LDW, K1, Wpad, biasp, bufA);
  rnn_scan     <<<1, 64, 0, stream>>>(bufA, W_hh1, bufB, TT);
  // Layer 2
  pad_weights  <<<(LDW * K1 + 255) / 256, 256, 0, stream>>>(W_ih2, b_ih2, b_hh2, H, K1, Wpad, biasp);
  wmma_pre_gemm<<<32, 128, 0, stream>>>(bufB, LDW, K1, Wpad, biasp, bufA);
  rnn_scan     <<<1, 64, 0, stream>>>(bufA, W_hh2, bufB, TT);
  // Head
  final_linear <<<(TT * 5 + 127) / 128, 128, 0, stream>>>(bufB, W_lin, b_lin, (float*)d_out);
}